// DeepPolyReLUTansformer_84662395339037
// MI455X (gfx1250) — compile-verified
//
#include <hip/hip_runtime.h>

// CDNA5 / gfx1250: wave32, V_WMMA_F32_16X16X4_F32 double-GEMV (fp32 throughout).
//
// Roofline: W is N*M fp32 = 256 MB; at 23.3 TB/s the floor is ~11 us. The kernel
// streams W exactly once via b128 loads (K-permuted WMMA slots keep A/B
// consistent per half-wave), the matrix pipe does all MACs, and a double-
// buffered register pipeline (kept small enough that the scheduler honors it)
// keeps ~12-24 loads in flight per wave.

typedef __attribute__((ext_vector_type(2))) float v2f;
typedef __attribute__((ext_vector_type(4))) float v4f;
typedef __attribute__((ext_vector_type(8))) float v8f;

#if defined(__HIP_DEVICE_COMPILE__)
#if !__has_builtin(__builtin_amdgcn_wmma_f32_16x16x4_f32)
#error "wmma_f32_16x16x4_f32 builtin not available on this toolchain (device pass)"
#endif
#endif

#define WAVES_PER_BLOCK 8
#define THREADS_PER_BLOCK (WAVES_PER_BLOCK * 32)
#define CH_DSTEPS 4                     // double-steps per chunk (K=8 each)
#define CH_K      (CH_DSTEPS * 8)       // 32 K per chunk

struct Chunk {                          // one prefetch chunk, in registers (48 dwords)
    v4f w[CH_DSTEPS];                   // W[n][k0+4h .. +3]
    v4f x[CH_DSTEPS];                   // PRE: sd pairs k,k+1 ; !PRE: plb[k..k+3]
    v4f y[CH_DSTEPS];                   // PRE: sd pairs k+2,k+3 ; !PRE: pub[k..k+3]
};

template<bool PRE>
__device__ __forceinline__ void load_chunk(Chunk& ch,
                                           const float* __restrict__ wptr,  // + kbase + 4h
                                           const float* __restrict__ sdp,   // + 2*(kbase+4h)
                                           const float* __restrict__ ap,    // + kbase + 4h
                                           const float* __restrict__ bp,    // + kbase + 4h
                                           int kc)
{
    #pragma unroll
    for (int i = 0; i < CH_DSTEPS; ++i) {
        const int k = kc + 8 * i;
        ch.w[i] = *(const v4f*)(wptr + k);
        if (PRE) {
            ch.x[i] = *(const v4f*)(sdp + 2 * k);        // s,d for k0, k0+1
            ch.y[i] = *(const v4f*)(sdp + 2 * k + 4);    // s,d for k0+2, k0+3
        } else {
            ch.x[i] = *(const v4f*)(ap + k);             // plb[k0..k0+3]
            ch.y[i] = *(const v4f*)(bp + k);             // pub[k0..k0+3]
        }
    }
}

// One K=8 double-step: two WMMA pairs fed from one W b128.
// Logical slot (2h+j) <-> physical k mapping is identical for A and B within
// each half-wave, so the accumulated dot product is exact under permutation.
template<bool PRE>
__device__ __forceinline__ void wmma_dstep(v4f w4, v4f x4, v4f y4, bool row0,
                                           v8f& aS0, v8f& aD0, v8f& aS1, v8f& aD1)
{
    v2f s1, d1, s2, d2;
    if (PRE) {
        s1.x = x4.x; d1.x = x4.y;  s1.y = x4.z; d1.y = x4.w;
        s2.x = y4.x; d2.x = y4.y;  s2.y = y4.z; d2.y = y4.w;
    } else {
        s1.x = (x4.x + y4.x) * 0.5f;  d1.x = (y4.x - x4.x) * 0.5f;
        s1.y = (x4.y + y4.y) * 0.5f;  d1.y = (y4.y - x4.y) * 0.5f;
        s2.x = (x4.z + y4.z) * 0.5f;  d2.x = (y4.z - x4.z) * 0.5f;
        s2.y = (x4.w + y4.w) * 0.5f;  d2.y = (y4.w - x4.w) * 0.5f;
    }
    v2f b1; b1.x = w4.x; b1.y = w4.y;
    v2f b2; b2.x = w4.z; b2.y = w4.w;
    v2f a1; a1.x = __builtin_fabsf(w4.x); a1.y = __builtin_fabsf(w4.y);
    v2f a2; a2.x = __builtin_fabsf(w4.z); a2.y = __builtin_fabsf(w4.w);

    v2f as1, ad1, as2, ad2;               // only A-row m==0 nonzero
    as1.x = row0 ? s1.x : 0.0f;  as1.y = row0 ? s1.y : 0.0f;
    ad1.x = row0 ? d1.x : 0.0f;  ad1.y = row0 ? d1.y : 0.0f;
    as2.x = row0 ? s2.x : 0.0f;  as2.y = row0 ? s2.y : 0.0f;
    ad2.x = row0 ? d2.x : 0.0f;  ad2.y = row0 ? d2.y : 0.0f;

    aS0 = __builtin_amdgcn_wmma_f32_16x16x4_f32(false, as1, false, b1, (short)0, aS0, false, false);
    aD0 = __builtin_amdgcn_wmma_f32_16x16x4_f32(false, ad1, false, a1, (short)0, aD0, false, false);
    aS1 = __builtin_amdgcn_wmma_f32_16x16x4_f32(false, as2, false, b2, (short)0, aS1, false, false);
    aD1 = __builtin_amdgcn_wmma_f32_16x16x4_f32(false, ad2, false, a2, (short)0, aD1, false, false);
}

template<bool PRE>
__device__ __forceinline__ void consume_chunk(const Chunk& ch, bool row0,
                                              v8f& aS0, v8f& aD0, v8f& aS1, v8f& aD1)
{
    #pragma unroll
    for (int i = 0; i < CH_DSTEPS; ++i)
        wmma_dstep<PRE>(ch.w[i], ch.x[i], ch.y[i], row0, aS0, aD0, aS1, aD1);
}

// Pre-kernel: interleaved sd[2i]=(plb+pub)/2, sd[2i+1]=(pub-plb)/2 into workspace.
__global__ void prep_sd_kernel(const float* __restrict__ last_bounds,
                               float* __restrict__ sd_out, int M)
{
    int i = blockIdx.x * blockDim.x + threadIdx.x;
    if (i < M) {
        float a = last_bounds[i];
        float b = last_bounds[M + i];
        sd_out[2 * i]     = (a + b) * 0.5f;
        sd_out[2 * i + 1] = (b - a) * 0.5f;
    }
}

template<bool PRE>
__global__ __launch_bounds__(THREADS_PER_BLOCK)
void deeppoly_relu_backsub_kernel(const float* __restrict__ bounds,      // [2,N]
                                  const float* __restrict__ W,           // [N,M]
                                  const float* __restrict__ bias,        // [N]
                                  const float* __restrict__ sv,          // PRE: sd[2M]; else plb
                                  const float* __restrict__ dv,          // PRE: unused; else pub
                                  const int*   __restrict__ back_sub,    // scalar
                                  float*       __restrict__ out,         // [2,N]
                                  int N, int M)
{
    __shared__ float red_low[WAVES_PER_BLOCK][16];
    __shared__ float red_up [WAVES_PER_BLOCK][16];

    const int tid  = threadIdx.x;
    const int wave = tid >> 5;
    const int lane = tid & 31;
    const int m    = lane & 15;
    const int h    = lane >> 4;
    const int r0   = blockIdx.x * 16;
    const int bs   = *back_sub;

    if (bs > 0) {
        const int kslice = M / WAVES_PER_BLOCK;   // 1024 for M=8192
        const int kbase  = wave * kslice;

        int rowB = r0 + m;
        if (rowB >= N) rowB = N - 1;

        const float* wrow = W + (size_t)rowB * (size_t)M + (size_t)kbase;
        // b128 path bases (physical offset 4h per lane)
        const float* wptr4 = wrow + 4 * h;
        const float* sdp4  = PRE ? (sv + 2 * (kbase + 4 * h)) : nullptr;
        const float* ap4   = sv + (kbase + 4 * h);
        const float* bp4   = dv + (kbase + 4 * h);
        // b64 tail bases (offset 2h per lane)
        const float* wptr2 = wrow + 2 * h;
        const float* sdp2  = PRE ? (sv + 2 * (kbase + 2 * h)) : nullptr;
        const float* ap2   = sv + (kbase + 2 * h);
        const float* bp2   = dv + (kbase + 2 * h);
        const bool   row0  = (m == 0);

        v8f aS0 = {}, aD0 = {}, aS1 = {}, aD1 = {};

        // ---- double-buffered pipelined main loop over 2*CH_K-aligned range ----
        const int ksl64 = kslice & ~(2 * CH_K - 1);   // multiple of 64
        int kdone = 0;
        if (ksl64 >= 2 * CH_K) {
            const int nch = ksl64 / CH_K;             // even, >= 2
            Chunk bufA, bufB;
            load_chunk<PRE>(bufA, wptr4, sdp4, ap4, bp4, 0);
            int c = 1;
            for (; c + 1 < nch; c += 2) {
                load_chunk<PRE>(bufB, wptr4, sdp4, ap4, bp4, c * CH_K);
                consume_chunk<PRE>(bufA, row0, aS0, aD0, aS1, aD1);
                load_chunk<PRE>(bufA, wptr4, sdp4, ap4, bp4, (c + 1) * CH_K);
                consume_chunk<PRE>(bufB, row0, aS0, aD0, aS1, aD1);
            }
            load_chunk<PRE>(bufB, wptr4, sdp4, ap4, bp4, (nch - 1) * CH_K);
            consume_chunk<PRE>(bufA, row0, aS0, aD0, aS1, aD1);
            consume_chunk<PRE>(bufB, row0, aS0, aD0, aS1, aD1);
            kdone = ksl64;
        }

        // ---- scalar tail, K=4 steps with the b64 (2h) mapping ----
        for (int kc = kdone; kc < kslice; kc += 4) {
            v2f wb = *(const v2f*)(wptr2 + kc);
            v2f s, d;
            if (PRE) {
                v4f sd = *(const v4f*)(sdp2 + 2 * kc);
                s.x = sd.x; d.x = sd.y; s.y = sd.z; d.y = sd.w;
            } else {
                v2f p = *(const v2f*)(ap2 + kc);
                v2f q = *(const v2f*)(bp2 + kc);
                s.x = (p.x + q.x) * 0.5f;  d.x = (q.x - p.x) * 0.5f;
                s.y = (p.y + q.y) * 0.5f;  d.y = (q.y - p.y) * 0.5f;
            }
            v2f wa; wa.x = __builtin_fabsf(wb.x); wa.y = __builtin_fabsf(wb.y);
            v2f as, ad;
            as.x = row0 ? s.x : 0.0f;  as.y = row0 ? s.y : 0.0f;
            ad.x = row0 ? d.x : 0.0f;  ad.y = row0 ? d.y : 0.0f;
            aS0 = __builtin_amdgcn_wmma_f32_16x16x4_f32(false, as, false, wb, (short)0, aS0, false, false);
            aD0 = __builtin_amdgcn_wmma_f32_16x16x4_f32(false, ad, false, wa, (short)0, aD0, false, false);
        }

        float ssum = aS0[0] + aS1[0];          // Sum_k W[n][k]*s[k]
        float dsum = aD0[0] + aD1[0];          // Sum_k |W[n][k]|*d[k]

        if (lane < 16) {
            red_low[wave][lane] = ssum - dsum;     // Wp@plb + Wn@pub
            red_up [wave][lane] = ssum + dsum;     // Wp@pub + Wn@plb
        }
    }

    __syncthreads();

    if (tid < 16) {
        int row = r0 + tid;
        if (row < N) {
            float lb = bounds[row];
            float ub = bounds[N + row];

            bool ind2 = (lb >= 0.0f);
            bool ind3 = (ub > 0.0f) && (lb < 0.0f);
            bool ind4 = (ub > -lb) && ind3;

            float diff  = ind3 ? (ub - lb) : 1.0f;
            float lmbda = ind2 ? 1.0f : (ind3 ? (ub / diff) : 0.0f);
            float betav = (ind2 || ind4) ? 1.0f : 0.0f;
            float mu    = ind3 ? ((-lb) * ub / diff) : 0.0f;

            float out_lb = (ind2 || ind4) ? lb : 0.0f;
            float out_ub = (ind2 || ind3) ? ub : 0.0f;

            if (bs > 0) {
                float lc = 0.0f, uc = 0.0f;
                #pragma unroll
                for (int w = 0; w < WAVES_PER_BLOCK; ++w) {
                    lc += red_low[w][tid];
                    uc += red_up [w][tid];
                }
                float b   = bias[row];
                float low = betav * (lc + b);
                float up  = lmbda * (uc + b) + mu;
                out_lb = fmaxf(out_lb, low);
                out_ub = fminf(out_ub, up);
            }

            out[row]     = out_lb;
            out[N + row] = out_ub;
        }
    }
}

extern "C" void kernel_launch(void* const* d_in, const int* in_sizes, int n_in,
                              void* d_out, int out_size, void* d_ws, size_t ws_size,
                              hipStream_t stream) {
    const float* bounds      = (const float*)d_in[0];
    const float* W           = (const float*)d_in[1];
    const float* bias        = (const float*)d_in[2];
    const float* last_bounds = (const float*)d_in[3];
    const int*   back_sub    = (const int*)d_in[4];
    float*       out         = (float*)d_out;

    const int N = in_sizes[0] / 2;
    const int M = in_sizes[3] / 2;

    const int grid = (N + 15) / 16;
    const bool pre = (ws_size >= (size_t)2 * (size_t)M * sizeof(float));

    if (pre) {
        float* sd_ws = (float*)d_ws;   // interleaved sd[2M]
        hipLaunchKernelGGL(prep_sd_kernel, dim3((M + 255) / 256), dim3(256), 0, stream,
                           last_bounds, sd_ws, M);
        hipLaunchKernelGGL((deeppoly_relu_backsub_kernel<true>),
                           dim3(grid), dim3(THREADS_PER_BLOCK), 0, stream,
                           bounds, W, bias, sd_ws, sd_ws, back_sub, out, N, M);
    } else {
        const float* plb = last_bounds;
        const float* pub = last_bounds + M;
        hipLaunchKernelGGL((deeppoly_relu_backsub_kernel<false>),
                           dim3(grid), dim3(THREADS_PER_BLOCK), 0, stream,
                           bounds, W, bias, plb, pub, back_sub, out, N, M);
    }
}